// SelfAttention_25228637897336
// MI455X (gfx1250) — compile-verified
//
#include <hip/hip_runtime.h>
#include <math.h>
#include <stdint.h>

// Problem constants (fixed by the reference)
#define BB 2
#define TT 2048
#define CC 1024
#define HH 16
#define DD 64
#define MM (BB*TT)          // 4096 rows in all GEMMs
#define BTC (BB*TT*CC)      // 4194304 elements per tensor

typedef float v2f __attribute__((ext_vector_type(2)));
typedef float v8f __attribute__((ext_vector_type(8)));
typedef int   gv4i __attribute__((vector_size(16)));   // builtin's pointee type

#define AS1 __attribute__((address_space(1)))
#define AS3 __attribute__((address_space(3)))

__device__ __forceinline__ v8f wmma_f32(v2f a, v2f b, v8f c) {
    return __builtin_amdgcn_wmma_f32_16x16x4_f32(
        /*neg_a=*/false, a, /*neg_b=*/false, b,
        /*c_mod=*/(short)0, c, /*reuse_a=*/false, /*reuse_b=*/false);
}

__device__ __forceinline__ v8f v8f_zero() {
    v8f z = {0.f,0.f,0.f,0.f,0.f,0.f,0.f,0.f};
    return z;
}

// Async 16B copy: global -> LDS, tracked by ASYNCcnt (no staging VGPRs).
__device__ __forceinline__ void async_load_b128(const float* g, float* l) {
#if defined(__HIP_DEVICE_COMPILE__)
#if __has_builtin(__builtin_amdgcn_global_load_async_to_lds_b128)
    __builtin_amdgcn_global_load_async_to_lds_b128(
        (AS1 gv4i*)(void*)const_cast<float*>(g),
        (AS3 gv4i*)(void*)l, /*offset=*/0, /*cpol=*/0);
#else
    const uint32_t loff = (uint32_t)(uintptr_t)(AS3 void*)(void*)l;
    asm volatile("global_load_async_to_lds_b128 %0, %1, off"
                 :: "v"(loff), "v"(g) : "memory");
#endif
#endif
}

template<int N>
__device__ __forceinline__ void wait_async() {
#if defined(__HIP_DEVICE_COMPILE__)
#if __has_builtin(__builtin_amdgcn_s_wait_asynccnt)
    __builtin_amdgcn_s_wait_asynccnt(N);
#else
    asm volatile("s_wait_asynccnt %0" :: "n"(N) : "memory");
#endif
#endif
}

// ---------------------------------------------------------------------------
// NT GEMM: out = X[M,K] @ W[N,K]^T + bias[N]
// HEAD=true  -> out in [B,H,T,D] layout (head-split), used for q/k/v
// HEAD=false -> out row-major [M,N], used for final projection
// Block: 256 threads (8 waves), tile 128(M) x 128(N), BK=16, double-buffered
// LDS fed by global_load_async_to_lds_b128 (prefetch tile kt+1 during tile kt).
// Wave (wm,wn) owns a 64x32 tile = 4x2 f32-WMMA accumulators.
// ---------------------------------------------------------------------------
template<bool HEAD>
__global__ __launch_bounds__(256)
void gemm_nt_kernel(const float* __restrict__ X, const float* __restrict__ W,
                    const float* __restrict__ bias, float* __restrict__ out)
{
    constexpr int BK = 16, LDK = 20;  // 80B rows: 16B aligned, conflict-free
    __shared__ float Xs[2][128][LDK];
    __shared__ float Ws[2][128][LDK];

    const int tid  = threadIdx.x;
    const int lane = tid & 31;
    const int wave = tid >> 5;
    const int wm   = wave >> 2;     // 0..1
    const int wn   = wave & 3;      // 0..3
    const int l16  = lane & 15;
    const int lhi  = lane >> 4;     // 0 or 1

    const int m0 = blockIdx.y * 128;
    const int n0 = blockIdx.x * 128;

    v8f acc[4][2];
    #pragma unroll
    for (int ms = 0; ms < 4; ++ms)
        #pragma unroll
        for (int ns = 0; ns < 2; ++ns) acc[ms][ns] = v8f_zero();

    // Per-thread staging coords: 128x16 floats = 512 float4 per tile, 2/thread
    const int i0  = tid;         // 0..255
    const int i1  = tid + 256;   // 256..511

    auto prefetch = [&](int buf, int k0) {
        {
            const int r = i0 >> 2, c4 = (i0 & 3) << 2;
            async_load_b128(X + (size_t)(m0 + r) * CC + k0 + c4, &Xs[buf][r][c4]);
            async_load_b128(W + (size_t)(n0 + r) * CC + k0 + c4, &Ws[buf][r][c4]);
        }
        {
            const int r = i1 >> 2, c4 = (i1 & 3) << 2;
            async_load_b128(X + (size_t)(m0 + r) * CC + k0 + c4, &Xs[buf][r][c4]);
            async_load_b128(W + (size_t)(n0 + r) * CC + k0 + c4, &Ws[buf][r][c4]);
        }
    };

    prefetch(0, 0);

    constexpr int NT = CC / BK;   // 64 K-tiles
    for (int kt = 0; kt < NT; ++kt) {
        const int buf = kt & 1;
        if (kt + 1 < NT) {
            prefetch(buf ^ 1, (kt + 1) * BK);  // overlaps with this tile's math
            wait_async<4>();                   // tile kt done; kt+1 in flight
        } else {
            wait_async<0>();
        }
        __syncthreads();

        #pragma unroll
        for (int kk = 0; kk < BK; kk += 4) {
            v2f a[4], b[2];
            #pragma unroll
            for (int ms = 0; ms < 4; ++ms)
                a[ms] = *(const v2f*)&Xs[buf][wm * 64 + ms * 16 + l16][kk + 2 * lhi];
            #pragma unroll
            for (int ns = 0; ns < 2; ++ns)
                b[ns] = *(const v2f*)&Ws[buf][wn * 32 + ns * 16 + l16][kk + 2 * lhi];
            #pragma unroll
            for (int ms = 0; ms < 4; ++ms)
                #pragma unroll
                for (int ns = 0; ns < 2; ++ns)
                    acc[ms][ns] = wmma_f32(a[ms], b[ns], acc[ms][ns]);
        }
        __syncthreads();   // all reads of this buffer done before it is re-filled
    }

    // Epilogue: bias add + store (C/D layout: row = r + 8*lhi, col = l16)
    #pragma unroll
    for (int ns = 0; ns < 2; ++ns) {
        const int n  = n0 + wn * 32 + ns * 16 + l16;
        const float bv = bias[n];
        #pragma unroll
        for (int ms = 0; ms < 4; ++ms) {
            #pragma unroll
            for (int r = 0; r < 8; ++r) {
                const int m = m0 + wm * 64 + ms * 16 + r + 8 * lhi;
                const float val = acc[ms][ns][r] + bv;
                size_t oidx;
                if (HEAD) {
                    const int bb = m >> 11;      // / T
                    const int t  = m & (TT - 1);
                    const int hh = n >> 6;       // / D
                    const int d  = n & (DD - 1);
                    oidx = (((size_t)bb * HH + hh) * TT + t) * DD + d;
                } else {
                    oidx = (size_t)m * CC + n;
                }
                out[oidx] = val;
            }
        }
    }
}

// ---------------------------------------------------------------------------
// Flash attention; mask(causal | blockdiag(64)) == full key blocks j <= i.
// Grid: (T/64, H, B). Block: 128 threads = 4 waves; wave w owns 16 query rows.
// k/v tiles arrive via async-to-LDS; v streams in during the S/softmax phase
// (wait_async<8> after issuing k then v). Scale folded into softmax FMA.
// ---------------------------------------------------------------------------
__global__ __launch_bounds__(128)
void attn_kernel(const float* __restrict__ Q, const float* __restrict__ Kc,
                 const float* __restrict__ Vc, float* __restrict__ Y)
{
    constexpr int LDD = 68;               // 64 + 4 pad: conflict-free, 16B rows
    __shared__ float qs [64][LDD];
    __shared__ float kps[64][LDD];        // k tile, reused for P after barrier
    __shared__ float vs [64][LDD];

    const int tid  = threadIdx.x;
    const int lane = tid & 31;
    const int wave = tid >> 5;            // 0..3
    const int l16  = lane & 15;
    const int lhi  = lane >> 4;

    const int qb = blockIdx.x;            // query block 0..31
    const int h  = blockIdx.y;
    const int b  = blockIdx.z;

    const size_t headBase = (((size_t)b * HH + h) * TT) * DD;
    const float* qg = Q + headBase + (size_t)qb * 64 * DD;

    // Async load of the q block (64x64): 8 x b128 per thread
    #pragma unroll
    for (int i = 0; i < 8; ++i) {
        const int idx4 = tid + i * 128;
        const int r    = idx4 >> 4;
        const int c4   = (idx4 & 15) << 2;
        async_load_b128(qg + r * DD + c4, &qs[r][c4]);
    }

    v8f o[4];
    #pragma unroll
    for (int dt = 0; dt < 4; ++dt) o[dt] = v8f_zero();
    float mstate[8], lstate[8];
    #pragma unroll
    for (int r = 0; r < 8; ++r) { mstate[r] = -INFINITY; lstate[r] = 0.f; }

    for (int j = 0; j <= qb; ++j) {
        __syncthreads();   // previous iteration's readers of kps/vs are done
        const float* kg = Kc + headBase + (size_t)j * 64 * DD;
        const float* vg = Vc + headBase + (size_t)j * 64 * DD;
        #pragma unroll
        for (int i = 0; i < 8; ++i) {      // k first ...
            const int idx4 = tid + i * 128;
            const int r    = idx4 >> 4;
            const int c4   = (idx4 & 15) << 2;
            async_load_b128(kg + r * DD + c4, &kps[r][c4]);
        }
        #pragma unroll
        for (int i = 0; i < 8; ++i) {      // ... then v (stays in flight)
            const int idx4 = tid + i * 128;
            const int r    = idx4 >> 4;
            const int c4   = (idx4 & 15) << 2;
            async_load_b128(vg + r * DD + c4, &vs[r][c4]);
        }
        wait_async<8>();   // q(+k) complete; the 8 v-loads still outstanding
        __syncthreads();

        // S = q_tile(16xD) @ k_tile(64xD)^T : 4 accumulators of 16x16
        v8f s[4];
        #pragma unroll
        for (int nt = 0; nt < 4; ++nt) s[nt] = v8f_zero();
        #pragma unroll
        for (int kk = 0; kk < DD; kk += 4) {
            const v2f a = *(const v2f*)&qs[wave * 16 + l16][kk + 2 * lhi];
            #pragma unroll
            for (int nt = 0; nt < 4; ++nt) {
                const v2f bf = *(const v2f*)&kps[nt * 16 + l16][kk + 2 * lhi];
                s[nt] = wmma_f32(a, bf, s[nt]);
            }
        }

        // Online softmax on scaled logits (scale = 1/sqrt(D) = 0.125, via FMA).
        // Physical row = r + 8*lhi; each row spans 16 lanes of a half-wave.
        #pragma unroll
        for (int r = 0; r < 8; ++r) {
            float mx = fmaxf(fmaxf(s[0][r], s[1][r]), fmaxf(s[2][r], s[3][r]));
            #pragma unroll
            for (int off = 1; off < 16; off <<= 1)
                mx = fmaxf(mx, __shfl_xor(mx, off, 32));
            mx *= 0.125f;                       // max commutes with pos. scale
            const float mnew = fmaxf(mstate[r], mx);
            const float corr = __expf(mstate[r] - mnew);
            mstate[r] = mnew;
            float sum = 0.f;
            #pragma unroll
            for (int nt = 0; nt < 4; ++nt) {
                const float p = __expf(fmaf(s[nt][r], 0.125f, -mnew));
                s[nt][r] = p;
                sum += p;
            }
            #pragma unroll
            for (int off = 1; off < 16; off <<= 1)
                sum += __shfl_xor(sum, off, 32);
            lstate[r] = lstate[r] * corr + sum;
            #pragma unroll
            for (int dt = 0; dt < 4; ++dt) o[dt][r] *= corr;
        }

        wait_async<0>();   // v tile fully landed (overlapped with S/softmax)
        __syncthreads();   // everyone done reading kps (k) -> reuse it for P
        #pragma unroll
        for (int nt = 0; nt < 4; ++nt)
            #pragma unroll
            for (int r = 0; r < 8; ++r)
                kps[wave * 16 + r + 8 * lhi][nt * 16 + l16] = s[nt][r];
        // same-wave LDS ops are in-order: A-frag reads below see our P writes

        // O += P(16x64) @ v_tile(64x64)
        #pragma unroll
        for (int kk = 0; kk < 64; kk += 4) {
            const v2f a = *(const v2f*)&kps[wave * 16 + l16][kk + 2 * lhi];
            #pragma unroll
            for (int dt = 0; dt < 4; ++dt) {
                v2f bf;
                bf.x = vs[kk + 2 * lhi + 0][dt * 16 + l16];
                bf.y = vs[kk + 2 * lhi + 1][dt * 16 + l16];
                o[dt] = wmma_f32(a, bf, o[dt]);
            }
        }
    }

    // Normalize and write Y in [B, T, H*D] so the projection GEMM reads row-major
    const int tbase = qb * 64 + wave * 16;
    #pragma unroll
    for (int r = 0; r < 8; ++r) {
        const float inv = 1.0f / lstate[r];
        const int t = tbase + r + 8 * lhi;
        #pragma unroll
        for (int dt = 0; dt < 4; ++dt) {
            const int d = dt * 16 + l16;
            Y[((size_t)b * TT + t) * CC + h * DD + d] = o[dt][r] * inv;
        }
    }
}

// ---------------------------------------------------------------------------
extern "C" void kernel_launch(void* const* d_in, const int* in_sizes, int n_in,
                              void* d_out, int out_size, void* d_ws, size_t ws_size,
                              hipStream_t stream)
{
    (void)in_sizes; (void)n_in; (void)out_size; (void)ws_size;
    const float* x  = (const float*)d_in[0];
    const float* Wq = (const float*)d_in[1];
    const float* bq = (const float*)d_in[2];
    const float* Wk = (const float*)d_in[3];
    const float* bk = (const float*)d_in[4];
    const float* Wv = (const float*)d_in[5];
    const float* bv = (const float*)d_in[6];
    const float* Wp = (const float*)d_in[7];
    const float* bp = (const float*)d_in[8];

    float* out  = (float*)d_out;
    float* yout = out;                 // [B,T,C]
    float* kout = out + (size_t)BTC;   // [B,H,T,D]
    float* vout = out + (size_t)2 * BTC;

    float* qws  = (float*)d_ws;            // [B,H,T,D] q
    float* yatt = qws + (size_t)BTC;       // [B,T,C]   attention output

    const dim3 gGrid(CC / 128, MM / 128);  // (8, 32)
    const dim3 gBlk(256);
    gemm_nt_kernel<true ><<<gGrid, gBlk, 0, stream>>>(x, Wq, bq, qws);
    gemm_nt_kernel<true ><<<gGrid, gBlk, 0, stream>>>(x, Wk, bk, kout);
    gemm_nt_kernel<true ><<<gGrid, gBlk, 0, stream>>>(x, Wv, bv, vout);

    const dim3 aGrid(TT / 64, HH, BB);     // (32, 16, 2)
    attn_kernel<<<aGrid, dim3(128), 0, stream>>>(qws, kout, vout, yatt);

    gemm_nt_kernel<false><<<gGrid, gBlk, 0, stream>>>(yatt, Wp, bp, yout);
}